// VisualMarkInjector_38525856645138
// MI455X (gfx1250) — compile-verified
//
#include <hip/hip_runtime.h>
#include <hip/hip_bf16.h>

// ---------- vector types ----------
typedef __attribute__((ext_vector_type(2))) float v2f;
typedef __attribute__((ext_vector_type(4))) float f4;
typedef __attribute__((ext_vector_type(8))) float v8f;
typedef __attribute__((ext_vector_type(4))) int   i4;

#define T_FRAMES 256
#define O_OBJ    64
#define K_MARKS  16
#define D_MODEL  768
#define D_OBJ    2048
#define HW       (224 * 224)
#define OMEGA    1e-6f

// =====================================================================
// Kernel 1/4: [16, Kdim] x [Kdim, N] + bias -> [16, N] via WMMA f32 16x16x4.
// One wave (32 threads) per 16x16 output tile; K loop in steps of 4.
//   A frag (16x4): lanes 0-15 hold rows M=l with K={k,k+1}; lanes 16-31 K={k+2,k+3}
//   B frag (4x16): v0 = B[k + 2*half][n0+l], v1 = B[k+1 + 2*half][n0+l]
//   D (16x16 f32, 8 VGPRs): vgpr r -> row r (lanes 0-15) / row r+8 (lanes 16-31), col l
// =====================================================================
__global__ void marks_gemm_wmma(const float* __restrict__ A,
                                const float* __restrict__ B,
                                const float* __restrict__ bias,
                                float* __restrict__ Out,
                                int Kdim, int N)
{
    const int lane   = threadIdx.x & 31;
    const int half   = lane >> 4;     // 0: lanes 0-15, 1: lanes 16-31
    const int l      = lane & 15;
    const int tile_n = blockIdx.x * 16;

    v8f acc = {};
    for (int k = 0; k < Kdim; k += 4) {
        const float* ap = A + (size_t)l * Kdim + k + half * 2;
        v2f a; a.x = ap[0]; a.y = ap[1];
        const float* bp = B + (size_t)(k + half * 2) * N + tile_n + l;
        v2f b; b.x = bp[0]; b.y = bp[N];
        acc = __builtin_amdgcn_wmma_f32_16x16x4_f32(
            /*neg_a=*/false, a, /*neg_b=*/false, b,
            /*c_mod=*/(short)0, acc, /*reuse_a=*/false, /*reuse_b=*/false);
    }

    const int   col = tile_n + l;
    const float bv  = bias[col];
#pragma unroll
    for (int r = 0; r < 8; ++r) {
        const int row = r + half * 8;
        Out[(size_t)row * N + col] = acc[r] + bv;
    }
}

// =====================================================================
// Kernel 2: obj_out = obj_feat + gamma * (entity>0 ? proj_obj[entity-1] : 0)
// 268 MB streamed -> non-temporal; proj_obj (128KB) stays cache-resident.
// grid = (D_OBJ/4/256, T*O), block = 256. One float4 per thread.
// =====================================================================
__global__ void obj_inject(const f4* __restrict__ obj4,
                           const float* __restrict__ proj_obj,
                           const int* __restrict__ ent,
                           const float* __restrict__ gamma_p,
                           f4* __restrict__ out4)
{
    const int    to = blockIdx.y;                                   // (t,o) flat
    const int    d4 = blockIdx.x * blockDim.x + threadIdx.x;        // 0..511
    const size_t i  = (size_t)to * (D_OBJ / 4) + d4;

    f4 v = __builtin_nontemporal_load(&obj4[i]);
    const int e = ent[to];                                          // uniform per block
    if (e > 0) {
        const float g  = gamma_p[0];
        const f4    pv = ((const f4*)proj_obj)[(size_t)(e - 1) * (D_OBJ / 4) + d4];
        v.x += g * pv.x; v.y += g * pv.y; v.z += g * pv.z; v.w += g * pv.w;
    }
    __builtin_nontemporal_store(v, &out4[i]);
}

// =====================================================================
// Kernel 3: per-frame label histogram of masks (labels 0..16, keep 1..16).
// One block (8 waves) per frame; per-wave private LDS histograms to cut
// atomic contention; int4 loads (50176 px = 12544 int4 per frame).
// =====================================================================
__global__ void mask_hist(const i4* __restrict__ masks4,
                          float* __restrict__ counts,    // [T,16]
                          float* __restrict__ weights)   // [T]
{
    __shared__ unsigned hist[8][20];                     // 17 bins, padded
    const int t    = blockIdx.x;
    const int wave = threadIdx.x >> 5;

    for (int i = threadIdx.x; i < 8 * 20; i += blockDim.x)
        ((unsigned*)hist)[i] = 0u;
    __syncthreads();

    const i4* p = masks4 + (size_t)t * (HW / 4);
    for (int i = threadIdx.x; i < HW / 4; i += 256) {
        const i4 m = p[i];
        atomicAdd(&hist[wave][m.x], 1u);
        atomicAdd(&hist[wave][m.y], 1u);
        atomicAdd(&hist[wave][m.z], 1u);
        atomicAdd(&hist[wave][m.w], 1u);
    }
    __syncthreads();

    if (threadIdx.x < K_MARKS) {
        unsigned s = 0;
#pragma unroll
        for (int w = 0; w < 8; ++w) s += hist[w][threadIdx.x + 1];  // skip label 0
        counts[t * K_MARKS + threadIdx.x] = (float)s;
    }
    if (threadIdx.x == 0) {
        unsigned tot = 0;
        for (int w = 0; w < 8; ++w)
            for (int k = 1; k <= K_MARKS; ++k) tot += hist[w][k];
        weights[t] = OMEGA + (float)tot;
    }
}

// =====================================================================
// Kernel 5: spatial = counts[256,16] @ proj_frame[16,768]  (WMMA, K=16)
//           frame_out = frame_feat + gamma * spatial / weight[row]
// grid = (768/16, 256/16), one wave per 16x16 tile.
// =====================================================================
__global__ void frame_inject_wmma(const float* __restrict__ counts,
                                  const float* __restrict__ proj_frame,
                                  const float* __restrict__ frame_feat,
                                  const float* __restrict__ weights,
                                  const float* __restrict__ gamma_p,
                                  float* __restrict__ out)
{
    const int lane   = threadIdx.x & 31;
    const int half   = lane >> 4;
    const int l      = lane & 15;
    const int tile_n = blockIdx.x * 16;
    const int tile_m = blockIdx.y * 16;
    const float g    = gamma_p[0];

    v8f acc = {};
#pragma unroll
    for (int k = 0; k < K_MARKS; k += 4) {
        const float* ap = counts + (size_t)(tile_m + l) * K_MARKS + k + half * 2;
        v2f a; a.x = ap[0]; a.y = ap[1];
        const float* bp = proj_frame + (size_t)(k + half * 2) * D_MODEL + tile_n + l;
        v2f b; b.x = bp[0]; b.y = bp[D_MODEL];
        acc = __builtin_amdgcn_wmma_f32_16x16x4_f32(
            false, a, false, b, (short)0, acc, false, false);
    }

    const int col = tile_n + l;
#pragma unroll
    for (int r = 0; r < 8; ++r) {
        const int row = tile_m + r + half * 8;
        out[(size_t)row * D_MODEL + col] =
            frame_feat[(size_t)row * D_MODEL + col] + g * acc[r] / weights[row];
    }
}

// =====================================================================
// Launch
// =====================================================================
extern "C" void kernel_launch(void* const* d_in, const int* in_sizes, int n_in,
                              void* d_out, int out_size, void* d_ws, size_t ws_size,
                              hipStream_t stream)
{
    const float* obj_feat   = (const float*)d_in[0];  // [T,O,D_OBJ]
    const float* frame_feat = (const float*)d_in[1];  // [T,D_MODEL]
    const float* mark_emb   = (const float*)d_in[2];  // [K,D_MODEL]
    const int*   obj_entity = (const int*)  d_in[3];  // [T,O]
    const int*   masks      = (const int*)  d_in[4];  // [T,H,W]
    const float* W_obj      = (const float*)d_in[5];  // [D_MODEL,D_OBJ]
    const float* b_obj      = (const float*)d_in[6];  // [D_OBJ]
    const float* W_frame    = (const float*)d_in[7];  // [D_MODEL,D_MODEL]
    const float* b_frame    = (const float*)d_in[8];  // [D_MODEL]
    const float* gamma      = (const float*)d_in[9];  // [1]

    float* obj_out   = (float*)d_out;                                  // T*O*D_OBJ
    float* frame_out = (float*)d_out + (size_t)T_FRAMES * O_OBJ * D_OBJ;

    char*  ws         = (char*)d_ws;
    float* proj_obj   = (float*)ws;                                              // 16*2048
    float* proj_frame = (float*)(ws + (size_t)K_MARKS * D_OBJ * 4);              // 16*768
    float* counts     = (float*)(ws + (size_t)K_MARKS * (D_OBJ + D_MODEL) * 4);  // 256*16
    float* weights    = counts + (size_t)T_FRAMES * K_MARKS;                     // 256

    // proj_obj = ME @ W_obj + b_obj        (128 tiles of 16x16, K=768)
    marks_gemm_wmma<<<dim3(D_OBJ / 16), 32, 0, stream>>>(
        mark_emb, W_obj, b_obj, proj_obj, D_MODEL, D_OBJ);

    // proj_frame = ME @ W_frame + b_frame  (48 tiles, K=768)
    marks_gemm_wmma<<<dim3(D_MODEL / 16), 32, 0, stream>>>(
        mark_emb, W_frame, b_frame, proj_frame, D_MODEL, D_MODEL);

    // per-frame label counts + weights
    mask_hist<<<dim3(T_FRAMES), 256, 0, stream>>>(
        (const i4*)masks, counts, weights);

    // obj_out: dominant streaming kernel
    obj_inject<<<dim3(D_OBJ / 4 / 256, T_FRAMES * O_OBJ), 256, 0, stream>>>(
        (const f4*)obj_feat, proj_obj, obj_entity, gamma, (f4*)obj_out);

    // frame_out: counts @ proj_frame with fused epilogue
    frame_inject_wmma<<<dim3(D_MODEL / 16, T_FRAMES / 16), 32, 0, stream>>>(
        counts, proj_frame, frame_feat, weights, gamma, frame_out);
}